// Aggregator_36636071035103
// MI455X (gfx1250) — compile-verified
//
#include <hip/hip_runtime.h>
#include <hip/hip_bf16.h>
#include <math.h>

#define N_USERS 100000
#define N_ITEMS 50000
#define N_ALL   150000
#define DIM     100
#define D4      25
#define NFAC    4
#define NREL    8
#define N_EDGES 1000000
#define N_P3    500000
#define N_P5    200000
#define NNZ     1000000
#define INV_T   200.0f   // 1/0.005

typedef __attribute__((ext_vector_type(2))) float v2f;
typedef __attribute__((ext_vector_type(8))) float v8f;
typedef __attribute__((ext_vector_type(4))) unsigned int u32x4;
typedef __attribute__((ext_vector_type(8))) int i32x8;
typedef __attribute__((ext_vector_type(4))) int i32x4;

#if __has_builtin(__builtin_amdgcn_tensor_load_to_lds)
#define USE_TDM 1
#else
#define USE_TDM 0
#endif

__device__ __forceinline__ void wave_lds_fence() {
    // intra-wave LDS ordering: DS ops complete in order; wait for all outstanding
    asm volatile("s_wait_dscnt 0" ::: "memory");
}

__device__ __forceinline__ void wait_tensor0() {
#if __has_builtin(__builtin_amdgcn_s_wait_tensorcnt)
    __builtin_amdgcn_s_wait_tensorcnt(0);
#else
    asm volatile("s_wait_tensorcnt 0" ::: "memory");
#endif
}

#if USE_TDM
// TDM 2-D tile load: 16 rows x DIM floats, row stride DIM floats, packed into
// LDS contiguously (Laddr increments continuously across rows -> [16][DIM]).
__device__ __forceinline__ void tdm_load_tile_16xD(const float* gsrc, void* ldst) {
    unsigned lds_addr = (unsigned)(unsigned long long)(uintptr_t)ldst;
    unsigned long long ga = (unsigned long long)(uintptr_t)gsrc;
    u32x4 g0;
    g0.x = 1u;                                   // count=1, user descriptor
    g0.y = (unsigned)__builtin_amdgcn_readfirstlane((int)lds_addr);
    g0.z = (unsigned)__builtin_amdgcn_readfirstlane((int)(unsigned)ga);
    g0.w = ((unsigned)__builtin_amdgcn_readfirstlane((int)(unsigned)(ga >> 32)))
           | (2u << 30);                         // type=2 ("image")
    i32x8 g1;
    g1[0] = 0x20000;                             // data_size=2 (4 bytes)
    g1[1] = (DIM << 16);                         // tensor_dim0[15:0] @ bits[63:48]
    g1[2] = (16 << 16);                          // tensor_dim1[15:0] @ bits[95:80]
    g1[3] = (DIM << 16);                         // tile_dim0 @ bits[127:112]
    g1[4] = 16;                                  // tile_dim1 @ bits[143:128]
    g1[5] = DIM;                                 // tensor_dim0_stride[31:0]
    g1[6] = 0;
    g1[7] = 0;
    i32x4 z4 = {0, 0, 0, 0};
    i32x8 z8 = {0, 0, 0, 0, 0, 0, 0, 0};
    // 6-arg toolchain variant: (g0, g1, g2, g3, extra, cpol)
    __builtin_amdgcn_tensor_load_to_lds(g0, g1, z4, z4, z8, 0);
}
#endif

// ---------------------------------------------------------------- zero fill
__global__ void zero_kernel(float4* __restrict__ p, int n4) {
    int i = blockIdx.x * blockDim.x + threadIdx.x;
    if (i < n4) p[i] = make_float4(0.f, 0.f, 0.f, 0.f);
}

// ------------------------------------------------- item scatter (sum part)
__global__ void item_scatter_kernel(const float4* __restrict__ all4,
                                    const int* __restrict__ head,
                                    const int* __restrict__ tail,
                                    float* __restrict__ item_out) {
    int i = blockIdx.x * blockDim.x + threadIdx.x;
    if (i >= N_EDGES * D4) return;
    int e = i / D4, c = i % D4;
    int h = head[e], t = tail[e];
    float4 v = all4[t * D4 + c];
    float* dst = item_out + h * DIM + c * 4;
    atomicAdd(dst + 0, v.x);
    atomicAdd(dst + 1, v.y);
    atomicAdd(dst + 2, v.z);
    atomicAdd(dst + 3, v.w);
}

__global__ void item_count_kernel(const int* __restrict__ head,
                                  float* __restrict__ counts) {
    int e = blockIdx.x * blockDim.x + threadIdx.x;
    if (e < N_EDGES) atomicAdd(&counts[head[e]], 1.0f);
}

__global__ void item_norm_kernel(float4* __restrict__ item4,
                                 const float* __restrict__ counts) {
    int i = blockIdx.x * blockDim.x + threadIdx.x;
    if (i >= N_ITEMS * D4) return;
    int row = i / D4;
    float inv = 1.0f / fmaxf(counts[row], 1.0f);
    float4 v = item4[i];
    v.x *= inv; v.y *= inv; v.z *= inv; v.w *= inv;
    item4[i] = v;
}

// ---------------------------------------- metapath -> softmax -> latent_agg
// One wave per 16-path tile:
//  tile[16][100] = mean over hops of gathered node embeddings (LDS)
//  score        = tile @ latent.T via 25x V_WMMA_F32_16X16X4_F32
//  softmax(score/T) over 4 factors, accumulate score.T @ tile into [4][100]
__global__ void path_latent_kernel(const float* __restrict__ all_embed,
                                   const int* __restrict__ nodes,
                                   int n_paths, int hops,
                                   const float* __restrict__ latent,
                                   float* __restrict__ latent_acc) {
    __shared__ float latLDS[16][DIM];    // rows 4..15 zero-padded (B operand)
    __shared__ float tile[4][16][DIM];
    __shared__ float scor[4][16][16];

    const int tid  = threadIdx.x;
    const int lane = tid & 31;
    const int wave = tid >> 5;
    const int half = lane >> 4;      // 0: lanes 0-15, 1: lanes 16-31
    const int l15  = lane & 15;

    for (int i = tid; i < 16 * DIM; i += blockDim.x) {
        int r = i / DIM, d = i % DIM;
        latLDS[r][d] = (r < NFAC) ? latent[r * DIM + d] : 0.f;
    }
    __syncthreads();

    float acc[4][NFAC];
    #pragma unroll
    for (int i = 0; i < 4; ++i)
        #pragma unroll
        for (int k = 0; k < NFAC; ++k) acc[i][k] = 0.f;

    const float inv_h = 1.0f / (float)hops;
    const int n_tiles = (n_paths + 15) / 16;
    const int waves_per_grid = gridDim.x * 4;

    for (int t = blockIdx.x * 4 + wave; t < n_tiles; t += waves_per_grid) {
        const int p0 = t * 16;
        // ---- build 16x100 path-embedding tile in LDS (coalesced gathers) ----
        for (int m = 0; m < 16; ++m) {
            int p = p0 + m;
            #pragma unroll
            for (int i = 0; i < 4; ++i) {
                int d = lane + 32 * i;
                if (d < DIM) {
                    float s = 0.f;
                    if (p < n_paths) {
                        for (int h = 0; h < hops; ++h) {
                            int nd = nodes[p * hops + h];
                            s += all_embed[nd * DIM + d];
                        }
                        s *= inv_h;
                    }
                    tile[wave][m][d] = s;
                }
            }
        }
        wave_lds_fence();

        // ---- scores = tile @ latent.T  (25 x wmma f32 16x16x4) ----
        v8f c = {};
        #pragma unroll
        for (int kk = 0; kk < DIM; kk += 4) {
            v2f a, b;
            a.x = tile[wave][l15][kk + 2 * half];
            a.y = tile[wave][l15][kk + 2 * half + 1];
            b.x = latLDS[l15][kk + 2 * half];
            b.y = latLDS[l15][kk + 2 * half + 1];
            c = __builtin_amdgcn_wmma_f32_16x16x4_f32(
                    false, a, false, b, (short)0, c, false, false);
        }
        #pragma unroll
        for (int r = 0; r < 8; ++r)
            scor[wave][r + 8 * half][l15] = c[r];
        wave_lds_fence();

        // ---- sharp softmax over 4 factors (one lane per path row) ----
        if (lane < 16) {
            float s0 = scor[wave][lane][0], s1 = scor[wave][lane][1];
            float s2 = scor[wave][lane][2], s3 = scor[wave][lane][3];
            float mx = fmaxf(fmaxf(s0, s1), fmaxf(s2, s3));
            float e0 = __expf((s0 - mx) * INV_T);
            float e1 = __expf((s1 - mx) * INV_T);
            float e2 = __expf((s2 - mx) * INV_T);
            float e3 = __expf((s3 - mx) * INV_T);
            float inv = 1.0f / (e0 + e1 + e2 + e3);
            scor[wave][lane][0] = e0 * inv;
            scor[wave][lane][1] = e1 * inv;
            scor[wave][lane][2] = e2 * inv;
            scor[wave][lane][3] = e3 * inv;
        }
        wave_lds_fence();

        // ---- accumulate score.T @ tile into registers ----
        #pragma unroll
        for (int i = 0; i < 4; ++i) {
            int d = lane + 32 * i;
            if (d < DIM) {
                for (int p = 0; p < 16; ++p) {
                    float tv = tile[wave][p][d];
                    acc[i][0] = fmaf(scor[wave][p][0], tv, acc[i][0]);
                    acc[i][1] = fmaf(scor[wave][p][1], tv, acc[i][1]);
                    acc[i][2] = fmaf(scor[wave][p][2], tv, acc[i][2]);
                    acc[i][3] = fmaf(scor[wave][p][3], tv, acc[i][3]);
                }
            }
        }
    }

    #pragma unroll
    for (int i = 0; i < 4; ++i) {
        int d = lane + 32 * i;
        if (d < DIM)
            for (int k = 0; k < NFAC; ++k)
                atomicAdd(&latent_acc[k * DIM + d], acc[i][k]);
    }
}

// -------------------------------------------------------------- disen weight
__global__ void disen_kernel(const float* __restrict__ att,
                             const float* __restrict__ weight,
                             const float* __restrict__ latent_acc,
                             float* __restrict__ dw_out) {
    __shared__ float soft[NFAC][NREL];
    __shared__ float red[128];
    int tid = threadIdx.x;
    if (tid < NFAC) {
        float mx = -1e30f;
        for (int r = 0; r < NREL; ++r) mx = fmaxf(mx, att[tid * NREL + r]);
        float sum = 0.f;
        for (int r = 0; r < NREL; ++r) {
            float e = __expf(att[tid * NREL + r] - mx);
            soft[tid][r] = e; sum += e;
        }
        float inv = 1.0f / sum;
        for (int r = 0; r < NREL; ++r) soft[tid][r] *= inv;
    }
    __syncthreads();
    for (int k = 0; k < NFAC; ++k) {
        float v = 0.f;
        if (tid < DIM) {
            float x = 0.f;
            for (int r = 0; r < NREL; ++r)
                x = fmaf(soft[k][r], weight[r * DIM + tid], x);
            v = 0.4f * x + 0.6f * latent_acc[k * DIM + tid];
        }
        red[tid] = v * v;
        __syncthreads();
        for (int s = 64; s > 0; s >>= 1) {
            if (tid < s) red[tid] += red[tid + s];
            __syncthreads();
        }
        float nrm = sqrtf(red[0]);
        if (tid < DIM) dw_out[k * DIM + tid] = v / fmaxf(nrm, 1e-12f);
        __syncthreads();
    }
}

// ----------------------------------------- user sparse mm (segment_sum part)
__global__ void user_sparse_kernel(const float4* __restrict__ item4,
                                   const int* __restrict__ rows,
                                   const int* __restrict__ cols,
                                   const float* __restrict__ vals,
                                   float* __restrict__ user_out) {
    int i = blockIdx.x * blockDim.x + threadIdx.x;
    if (i >= NNZ * D4) return;
    int nz = i / D4, c = i % D4;
    int r = rows[nz];
    float vl = vals[nz];
    float4 v = item4[cols[nz] * D4 + c];
    float* dst = user_out + r * DIM + c * 4;
    atomicAdd(dst + 0, v.x * vl);
    atomicAdd(dst + 1, v.y * vl);
    atomicAdd(dst + 2, v.z * vl);
    atomicAdd(dst + 3, v.w * vl);
}

// --------------------- user score softmax (WMMA) + factor modulation of agg
// A-tile (16 users x 100 dims) staged LDS-side via the Tensor Data Mover.
__global__ void user_mod_kernel(const float* __restrict__ user_emb,
                                const float* __restrict__ latent,
                                const float* __restrict__ dw,
                                float* __restrict__ user_out) {
    __shared__ float latLDS[16][DIM];    // rows 4..15 zero-padded (B operand)
    __shared__ float dwLDS[NFAC][DIM];
    __shared__ float tileU[4][16][DIM];
    __shared__ float scor[4][16][16];

    const int tid  = threadIdx.x;
    const int lane = tid & 31;
    const int wave = tid >> 5;
    const int half = lane >> 4;
    const int l15  = lane & 15;

    for (int i = tid; i < 16 * DIM; i += blockDim.x) {
        int r = i / DIM, d = i % DIM;
        latLDS[r][d] = (r < NFAC) ? latent[r * DIM + d] : 0.f;
        if (r < NFAC) dwLDS[r][d] = dw[r * DIM + d];
    }
    __syncthreads();

    const int n_tiles = N_USERS / 16;   // 6250 exact
    const int waves_per_grid = gridDim.x * 4;

    for (int t = blockIdx.x * 4 + wave; t < n_tiles; t += waves_per_grid) {
        const int base = t * 16;

        // ---- stage user_emb tile [16][100] into LDS ----
        wave_lds_fence();   // previous-iteration LDS reads complete before overwrite
#if USE_TDM
        tdm_load_tile_16xD(user_emb + (size_t)base * DIM, &tileU[wave][0][0]);
        wait_tensor0();
#else
        for (int m = 0; m < 16; ++m) {
            #pragma unroll
            for (int i = 0; i < 4; ++i) {
                int d = lane + 32 * i;
                if (d < DIM) tileU[wave][m][d] = user_emb[(base + m) * DIM + d];
            }
        }
        wave_lds_fence();
#endif

        // ---- scores = tile @ latent.T (25 x wmma f32 16x16x4) ----
        v8f c = {};
        #pragma unroll
        for (int kk = 0; kk < DIM; kk += 4) {
            v2f a, b;
            a.x = tileU[wave][l15][kk + 2 * half];
            a.y = tileU[wave][l15][kk + 2 * half + 1];
            b.x = latLDS[l15][kk + 2 * half];
            b.y = latLDS[l15][kk + 2 * half + 1];
            c = __builtin_amdgcn_wmma_f32_16x16x4_f32(
                    false, a, false, b, (short)0, c, false, false);
        }
        #pragma unroll
        for (int r = 0; r < 8; ++r)
            scor[wave][r + 8 * half][l15] = c[r];
        wave_lds_fence();

        if (lane < 16) {
            float s0 = scor[wave][lane][0], s1 = scor[wave][lane][1];
            float s2 = scor[wave][lane][2], s3 = scor[wave][lane][3];
            float mx = fmaxf(fmaxf(s0, s1), fmaxf(s2, s3));
            float e0 = __expf(s0 - mx), e1 = __expf(s1 - mx);
            float e2 = __expf(s2 - mx), e3 = __expf(s3 - mx);
            float inv = 1.0f / (e0 + e1 + e2 + e3);
            scor[wave][lane][0] = e0 * inv;
            scor[wave][lane][1] = e1 * inv;
            scor[wave][lane][2] = e2 * inv;
            scor[wave][lane][3] = e3 * inv;
        }
        wave_lds_fence();

        // user_agg *= (1 + sum_k score[k]*disen_weight[k][d])
        for (int p = 0; p < 16; ++p) {
            int row = base + p;
            float s0 = scor[wave][p][0], s1 = scor[wave][p][1];
            float s2 = scor[wave][p][2], s3 = scor[wave][p][3];
            #pragma unroll
            for (int i = 0; i < 4; ++i) {
                int d = lane + 32 * i;
                if (d < DIM) {
                    float mod = s0 * dwLDS[0][d] + s1 * dwLDS[1][d]
                              + s2 * dwLDS[2][d] + s3 * dwLDS[3][d];
                    float v = user_out[row * DIM + d];
                    user_out[row * DIM + d] = fmaf(v, mod, v);
                }
            }
        }
    }
}

// ---------------------------------------------------------------------------
extern "C" void kernel_launch(void* const* d_in, const int* in_sizes, int n_in,
                              void* d_out, int out_size, void* d_ws, size_t ws_size,
                              hipStream_t stream) {
    const float* user_emb  = (const float*)d_in[0];
    const float* item_emb  = (const float*)d_in[1];
    const float* all_embed = (const float*)d_in[2];
    const float* latent    = (const float*)d_in[3];
    const float* weight    = (const float*)d_in[4];
    const float* att       = (const float*)d_in[5];
    const int*   head      = (const int*)d_in[6];
    const int*   tail      = (const int*)d_in[7];
    const int*   p3nodes   = (const int*)d_in[8];
    const int*   p5nodes   = (const int*)d_in[9];
    const int*   mrows     = (const int*)d_in[10];
    const int*   mcols     = (const int*)d_in[11];
    const float* mvals     = (const float*)d_in[12];

    float* out_user = (float*)d_out;                    // [N_USERS, DIM]
    float* out_item = out_user + (size_t)N_USERS * DIM; // [N_ITEMS, DIM]

    float* ws_f    = (float*)d_ws;
    float* counts  = ws_f;                 // [N_ITEMS]
    float* lat_acc = ws_f + N_ITEMS;       // [NFAC*DIM] = 400
    float* dw      = lat_acc + NFAC * DIM; // [NFAC*DIM] = 400

    const int B = 256;

    // zero outputs + accumulators (poison-safe, deterministic per call)
    {
        int n4 = (N_USERS + N_ITEMS) * DIM / 4;
        zero_kernel<<<(n4 + B - 1) / B, B, 0, stream>>>((float4*)d_out, n4);
        int w4 = (N_ITEMS + 2 * NFAC * DIM + 3) / 4;
        zero_kernel<<<(w4 + B - 1) / B, B, 0, stream>>>((float4*)ws_f, w4);
    }

    // ---- item aggregation (scatter mean) ----
    {
        int work = N_EDGES * D4;
        item_scatter_kernel<<<(work + B - 1) / B, B, 0, stream>>>(
            (const float4*)all_embed, head, tail, out_item);
        item_count_kernel<<<(N_EDGES + B - 1) / B, B, 0, stream>>>(head, counts);
        int nw = N_ITEMS * D4;
        item_norm_kernel<<<(nw + B - 1) / B, B, 0, stream>>>(
            (float4*)out_item, counts);
    }

    // ---- metapath -> latent_agg (WMMA score matmul + sharp softmax) ----
    path_latent_kernel<<<1024, 128, 0, stream>>>(
        all_embed, p3nodes, N_P3, 3, latent, lat_acc);
    path_latent_kernel<<<1024, 128, 0, stream>>>(
        all_embed, p5nodes, N_P5, 5, latent, lat_acc);

    // ---- disentangled factor weights ----
    disen_kernel<<<1, 128, 0, stream>>>(att, weight, lat_acc, dw);

    // ---- user aggregation: sparse mm then WMMA score + modulation ----
    {
        int work = NNZ * D4;
        user_sparse_kernel<<<(work + B - 1) / B, B, 0, stream>>>(
            (const float4*)item_emb, mrows, mcols, mvals, out_user);
    }
    user_mod_kernel<<<1563, 128, 0, stream>>>(user_emb, latent, dw, out_user);
}